// GNNModel_90108413870647
// MI455X (gfx1250) — compile-verified
//
#include <hip/hip_runtime.h>

#define N_NODES 50000
#define N_EDGES 800000
#define DIN     128
#define DH      64
#define NB      64

typedef __attribute__((ext_vector_type(2))) float v2f;
typedef __attribute__((ext_vector_type(8))) float v8f;

// ---------------- degree / normalization ----------------
__global__ void k_deg_init(float* __restrict__ deg) {
  int i = blockIdx.x * blockDim.x + threadIdx.x;
  if (i < N_NODES) deg[i] = 1.0f;              // self-loop
}

__global__ void k_deg_count(const int* __restrict__ dst, float* __restrict__ deg) {
  int i = blockIdx.x * blockDim.x + threadIdx.x;
  if (i < N_EDGES) unsafeAtomicAdd(&deg[dst[i]], 1.0f);
}

__global__ void k_deg_rsqrt(float* __restrict__ deg) {
  int i = blockIdx.x * blockDim.x + threadIdx.x;
  if (i < N_NODES) deg[i] = rsqrtf(deg[i]);    // deg >= 1, safe
}

__global__ void k_zero(float* __restrict__ p, int n) {
  int i = blockIdx.x * blockDim.x + threadIdx.x;
  if (i < n) p[i] = 0.0f;
}

// ---------------- WMMA GEMM: H[N x DH] = X[N x K] @ W[K x DH] ----------------
// Block: 128 threads = 4 waves; each wave owns one 16-col tile of DH=64.
// Epilogue also writes AGG = H * dinv^2 (self-loop term of GCN aggregation).
template <int K>
__launch_bounds__(128)
__global__ void k_gemm_wmma(const float* __restrict__ X, const float* __restrict__ W,
                            const float* __restrict__ dinv,
                            float* __restrict__ H, float* __restrict__ AGG) {
  constexpr int KP = K + 4;                    // pad LDS rows: avoid 16-way bank conflicts
  __shared__ float sA[16 * KP];

  const int m0   = blockIdx.x * 16;
  const int tid  = threadIdx.x;
  const int wave = tid >> 5;
  const int lane = tid & 31;

  // cooperative load of 16 x K A-tile (row-major) into padded LDS, float4 at a time
  const float4* __restrict__ g4 = (const float4*)(X + (size_t)m0 * K);
  for (int i4 = tid; i4 < (16 * K) / 4; i4 += 128) {
    int row = i4 / (K / 4);
    int c4  = i4 % (K / 4);
    ((float4*)(sA + row * KP))[c4] = g4[row * (K / 4) + c4];
  }
  __syncthreads();

  const int n0 = wave * 16;
  const int nm = lane & 15;          // A row m / B,D col n
  const int kh = (lane >> 4) * 2;    // K sub-offset within group of 4

  v8f acc = {};
  #pragma unroll 4
  for (int k = 0; k < K; k += 4) {
    v2f a, b;
    a.x = sA[nm * KP + k + kh + 0];
    a.y = sA[nm * KP + k + kh + 1];
    b.x = W[(k + kh + 0) * DH + n0 + nm];
    b.y = W[(k + kh + 1) * DH + n0 + nm];
    acc = __builtin_amdgcn_wmma_f32_16x16x4_f32(
        /*neg_a=*/false, a, /*neg_b=*/false, b,
        /*c_mod=*/(short)0, acc, /*reuse_a=*/false, /*reuse_b=*/false);
  }

  // C/D layout: lane n = lane&15; VGPR r -> m = r + (lane>>4)*8
  const int mh = (lane >> 4) * 8;
  #pragma unroll
  for (int r = 0; r < 8; ++r) {
    int m = m0 + mh + r;
    float h  = acc[r];
    float di = dinv[m];
    size_t off = (size_t)m * DH + n0 + nm;
    H[off]   = h;
    AGG[off] = h * (di * di);
  }
}

// ---------------- edge scatter: AGG[dst] += H[src] * dinv[src]*dinv[dst] ----------------
// one thread per (edge, float4 feature chunk): E*16 threads
__global__ void k_edge_scatter(const int* __restrict__ src, const int* __restrict__ dst,
                               const float* __restrict__ dinv,
                               const float* __restrict__ H, float* __restrict__ AGG) {
  int i = blockIdx.x * blockDim.x + threadIdx.x;
  if (i >= N_EDGES * 16) return;
  int e = i >> 4;
  int c = (i & 15) << 2;
  int s = src[e], d = dst[e];
  float w = dinv[s] * dinv[d];
  const float4 hv = *(const float4*)(H + (size_t)s * DH + c);
  float* ap = AGG + (size_t)d * DH + c;
  unsafeAtomicAdd(ap + 0, hv.x * w);
  unsafeAtomicAdd(ap + 1, hv.y * w);
  unsafeAtomicAdd(ap + 2, hv.z * w);
  unsafeAtomicAdd(ap + 3, hv.w * w);
}

// ---------------- relu(agg + bias) ----------------
__global__ void k_finalize(const float* __restrict__ AGG, const float* __restrict__ bias,
                           float* __restrict__ OUT) {
  int i = blockIdx.x * blockDim.x + threadIdx.x;
  if (i >= N_NODES * DH) return;
  OUT[i] = fmaxf(AGG[i] + bias[i & (DH - 1)], 0.0f);
}

// layer-3 finalize fused with segment-sum pooling
__global__ void k_finalize_pool(const float* __restrict__ AGG, const float* __restrict__ bias,
                                const int* __restrict__ batch,
                                float* __restrict__ psum, float* __restrict__ pcnt) {
  int i = blockIdx.x * blockDim.x + threadIdx.x;
  if (i >= N_NODES * DH) return;
  int node = i >> 6;
  int f    = i & (DH - 1);
  float v  = fmaxf(AGG[i] + bias[f], 0.0f);
  int b = batch[node];
  unsafeAtomicAdd(&psum[b * DH + f], v);
  if (f == 0) unsafeAtomicAdd(&pcnt[b], 1.0f);
}

// ---------------- mean pool + FC (DH -> 1) ----------------
__global__ void k_pool_fc(const float* __restrict__ psum, const float* __restrict__ pcnt,
                          const float* __restrict__ Wfc, const float* __restrict__ bfc,
                          float* __restrict__ out) {
  __shared__ float red[DH];
  int b = blockIdx.x, f = threadIdx.x;
  float cnt = fmaxf(pcnt[b], 1.0f);
  red[f] = (psum[b * DH + f] / cnt) * Wfc[f];
  __syncthreads();
  for (int s = DH / 2; s > 0; s >>= 1) {
    if (f < s) red[f] += red[f + s];
    __syncthreads();
  }
  if (f == 0) out[b] = red[0] + bfc[0];
}

// ---------------- launcher ----------------
extern "C" void kernel_launch(void* const* d_in, const int* in_sizes, int n_in,
                              void* d_out, int out_size, void* d_ws, size_t ws_size,
                              hipStream_t stream) {
  const float* x    = (const float*)d_in[0];
  const int*   edge = (const int*)d_in[1];
  const int*   batch= (const int*)d_in[2];
  const float* W1   = (const float*)d_in[3];
  const float* b1   = (const float*)d_in[4];
  const float* W2   = (const float*)d_in[5];
  const float* b2   = (const float*)d_in[6];
  const float* W3   = (const float*)d_in[7];
  const float* b3   = (const float*)d_in[8];
  const float* Wfc  = (const float*)d_in[9];
  const float* bfc  = (const float*)d_in[10];
  float* out = (float*)d_out;

  const int* src = edge;
  const int* dst = edge + N_EDGES;

  char* ws = (char*)d_ws;
  size_t off = 0;
  auto alloc = [&](size_t bytes) -> void* {
    void* p = ws + off;
    off = (off + bytes + 255) & ~(size_t)255;
    return p;
  };
  float* dinv   = (float*)alloc((size_t)N_NODES * 4);
  float* bufH   = (float*)alloc((size_t)N_NODES * DH * 4);
  float* bufAgg = (float*)alloc((size_t)N_NODES * DH * 4);
  float* bufOut = (float*)alloc((size_t)N_NODES * DH * 4);
  float* psum   = (float*)alloc((size_t)NB * DH * 4);
  float* pcnt   = (float*)alloc((size_t)NB * 4);

  const int TB = 256;
  const int gN  = (N_NODES + TB - 1) / TB;
  const int gE  = (N_EDGES + TB - 1) / TB;
  const int gE16 = (N_EDGES * 16 + TB - 1) / TB;
  const int gNF = (N_NODES * DH + TB - 1) / TB;
  const int gTiles = N_NODES / 16;   // 3125, exact

  // normalization coefficients
  k_deg_init <<<gN, TB, 0, stream>>>(dinv);
  k_deg_count<<<gE, TB, 0, stream>>>(dst, dinv);
  k_deg_rsqrt<<<gN, TB, 0, stream>>>(dinv);
  k_zero<<<(NB * DH + TB - 1) / TB, TB, 0, stream>>>(psum, NB * DH);
  k_zero<<<1, NB, 0, stream>>>(pcnt, NB);

  // layer 1
  k_gemm_wmma<DIN><<<gTiles, 128, 0, stream>>>(x, W1, dinv, bufH, bufAgg);
  k_edge_scatter<<<gE16, TB, 0, stream>>>(src, dst, dinv, bufH, bufAgg);
  k_finalize<<<gNF, TB, 0, stream>>>(bufAgg, b1, bufOut);
  // layer 2
  k_gemm_wmma<DH><<<gTiles, 128, 0, stream>>>(bufOut, W2, dinv, bufH, bufAgg);
  k_edge_scatter<<<gE16, TB, 0, stream>>>(src, dst, dinv, bufH, bufAgg);
  k_finalize<<<gNF, TB, 0, stream>>>(bufAgg, b2, bufOut);
  // layer 3 (finalize fused with pooling accumulation)
  k_gemm_wmma<DH><<<gTiles, 128, 0, stream>>>(bufOut, W3, dinv, bufH, bufAgg);
  k_edge_scatter<<<gE16, TB, 0, stream>>>(src, dst, dinv, bufH, bufAgg);
  k_finalize_pool<<<gNF, TB, 0, stream>>>(bufAgg, b3, batch, psum, pcnt);

  // mean pool + FC
  k_pool_fc<<<NB, DH, 0, stream>>>(psum, pcnt, Wfc, bfc, out);
}